// LuongAttnDecoderRNN_61744449847943
// MI455X (gfx1250) — compile-verified
//
#include <hip/hip_runtime.h>
#include <hip/hip_bf16.h>
#include <math.h>

// Problem dims
#define B_  64
#define L_  2048
#define H_  1024
#define V_  32000
#define P_  128

#define NC_     8               // L chunks for fused attention
#define CHUNK_  (L_ / NC_)      // 256 l's per block; 32 per wave

typedef __attribute__((ext_vector_type(2))) float v2f;
typedef __attribute__((ext_vector_type(8))) float v8f;

// ---------------------------------------------------------------------------
// Generic WMMA-f32 GEMM:  out[M,N] = act( A[M,K] * W[N,K]^T + bias[N] )
// M = 64 fixed (4 M-tiles, one per wave of a 128-thread block).
// grid.x = N/16, block = 128. Uses V_WMMA_F32_16X16X4_F32.
// Two independent accumulator chains (even/odd k-quads) so consecutive
// WMMAs have no RAW dependency; K must be a multiple of 8 (1024/2048/1152 ok).
// ---------------------------------------------------------------------------
__global__ __launch_bounds__(128) void gemm_wmma_f32(
    const float* __restrict__ A, int lda,
    const float* __restrict__ W, int ldw,
    const float* __restrict__ bias,
    float* __restrict__ out, int ldo,
    int K, int act_tanh)
{
    const int lane = threadIdx.x & 31;
    const int mt   = threadIdx.x >> 5;        // 0..3 -> m-tile
    const int n0   = blockIdx.x * 16;
    const int m0   = mt * 16;
    const int half = lane >> 4;               // 0 or 1
    const int l15  = lane & 15;

    const float* Ap = A + (size_t)(m0 + l15) * lda + 2 * half;
    const float* Wp = W + (size_t)(n0 + l15) * ldw + 2 * half;

    v8f acc0 = {};
    v8f acc1 = {};
    for (int k = 0; k < K; k += 8) {
        v2f a0 = *(const v2f*)(Ap + k);
        v2f b0 = *(const v2f*)(Wp + k);
        v2f a1 = *(const v2f*)(Ap + k + 4);
        v2f b1 = *(const v2f*)(Wp + k + 4);
        __builtin_prefetch(Wp + k + 64, 0, 0);   // global_prefetch_b8
        acc0 = __builtin_amdgcn_wmma_f32_16x16x4_f32(
                   false, a0, false, b0, (short)0, acc0, false, false);
        acc1 = __builtin_amdgcn_wmma_f32_16x16x4_f32(
                   false, a1, false, b1, (short)0, acc1, false, false);
    }

    const int   ncol = n0 + l15;
    const float bs   = bias ? bias[ncol] : 0.0f;
    const int   mrow = m0 + half * 8;
    #pragma unroll
    for (int r = 0; r < 8; ++r) {
        float v = acc0[r] + acc1[r] + bs;
        if (act_tanh) v = tanhf(v);
        out[(size_t)(mrow + r) * ldo + ncol] = v;
    }
}

// ---------------------------------------------------------------------------
// Embedding gather: xbuf[b,h] = emb[seq[b], h].  grid (B, H/256), block 256.
// ---------------------------------------------------------------------------
__global__ __launch_bounds__(256) void k_embed(
    const int* __restrict__ seq, const float* __restrict__ emb,
    float* __restrict__ xbuf)
{
    const int b = blockIdx.x;
    const int h = blockIdx.y * 256 + threadIdx.x;
    const int tok = seq[b];
    xbuf[(size_t)b * H_ + h] = emb[(size_t)tok * H_ + h];
}

// ---------------------------------------------------------------------------
// GRU gate math.  gi,gh: [B,3H].  grid (B, H/256), block 256.
// ---------------------------------------------------------------------------
__global__ __launch_bounds__(256) void k_gru_gate(
    const float* __restrict__ gi, const float* __restrict__ gh,
    const float* __restrict__ hprev,
    float* __restrict__ hnew, float* __restrict__ concatbuf)
{
    const int b = blockIdx.x;
    const int h = blockIdx.y * 256 + threadIdx.x;
    const size_t o3 = (size_t)b * (3 * H_);
    const float ir = gi[o3 + h], iz = gi[o3 + H_ + h], in_ = gi[o3 + 2 * H_ + h];
    const float hr = gh[o3 + h], hz = gh[o3 + H_ + h], hn  = gh[o3 + 2 * H_ + h];
    const float r = 1.0f / (1.0f + __expf(-(ir + hr)));
    const float z = 1.0f / (1.0f + __expf(-(iz + hz)));
    const float n = tanhf(in_ + r * hn);
    const float hp = hprev[(size_t)b * H_ + h];
    const float hv = (1.0f - z) * n + z * hp;
    hnew[(size_t)b * H_ + h] = hv;
    concatbuf[(size_t)b * (2 * H_) + h] = hv;
}

// ---------------------------------------------------------------------------
// Fused attention (flash-style, single pass over encoder_outputs):
// grid (B, NC), block 256 (8 waves). Wave w handles 32 consecutive l's.
// Per l: load enc row as 8 x float4 per lane (reused for dot AND context),
// shfl-reduce the energy, online-softmax rescale the ctx accumulator.
// Waves merge in LDS (ds_add_f32); block emits per-chunk partial (ctx,m,s).
// Raw energies are also stored for the attention-weights output pass.
// ---------------------------------------------------------------------------
__global__ __launch_bounds__(256) void k_attn_fused(
    const float* __restrict__ hnew, const float* __restrict__ enc,
    float* __restrict__ energies,      // [B,L] raw
    float* __restrict__ part_ctx,      // [B,NC,H]
    float* __restrict__ part_ms)       // [B,NC,2] (m, s)
{
    __shared__ float ctx_sh[H_];
    __shared__ float m_sh[8], s_sh[8];

    const int b    = blockIdx.x;
    const int c    = blockIdx.y;
    const int w    = threadIdx.x >> 5;
    const int lane = threadIdx.x & 31;

    for (int h = threadIdx.x; h < H_; h += 256) ctx_sh[h] = 0.0f;
    __syncthreads();

    // h_new row, resident in registers (reused across all 32 l's)
    const float4* hp = (const float4*)(hnew + (size_t)b * H_);
    float4 hreg[8];
    #pragma unroll
    for (int j = 0; j < 8; ++j) hreg[j] = hp[lane + 32 * j];

    float m_w = -3.4e38f, s_w = 0.0f;
    float4 ctx[8];
    #pragma unroll
    for (int j = 0; j < 8; ++j) ctx[j] = make_float4(0.f, 0.f, 0.f, 0.f);

    const int lbase = c * CHUNK_ + w * (CHUNK_ / 8);
    for (int li = 0; li < CHUNK_ / 8; ++li) {
        const int l = lbase + li;
        const float4* ep = (const float4*)(enc + ((size_t)l * B_ + b) * H_);
        float4 ev[8];
        float  e = 0.0f;
        #pragma unroll
        for (int j = 0; j < 8; ++j) {
            ev[j] = ep[lane + 32 * j];
            e += hreg[j].x * ev[j].x + hreg[j].y * ev[j].y
               + hreg[j].z * ev[j].z + hreg[j].w * ev[j].w;
        }
        #pragma unroll
        for (int mk = 16; mk >= 1; mk >>= 1) e += __shfl_xor(e, mk, 32);

        if (lane == 0) energies[(size_t)b * L_ + l] = e;

        const float m_new = fmaxf(m_w, e);
        const float scale = __expf(m_w - m_new);     // 0 on first iter
        const float wgt   = __expf(e - m_new);
        s_w = s_w * scale + wgt;
        #pragma unroll
        for (int j = 0; j < 8; ++j) {
            ctx[j].x = ctx[j].x * scale + wgt * ev[j].x;
            ctx[j].y = ctx[j].y * scale + wgt * ev[j].y;
            ctx[j].z = ctx[j].z * scale + wgt * ev[j].z;
            ctx[j].w = ctx[j].w * scale + wgt * ev[j].w;
        }
        m_w = m_new;
    }

    // ---- merge 8 waves within the block ----
    if (lane == 0) m_sh[w] = m_w;
    __syncthreads();
    float m_b = m_sh[0];
    #pragma unroll
    for (int i = 1; i < 8; ++i) m_b = fmaxf(m_b, m_sh[i]);
    const float f = __expf(m_w - m_b);
    if (lane == 0) s_sh[w] = s_w * f;
    #pragma unroll
    for (int j = 0; j < 8; ++j) {
        const int h = lane * 4 + 128 * j;
        atomicAdd(&ctx_sh[h + 0], ctx[j].x * f);    // ds_add_f32
        atomicAdd(&ctx_sh[h + 1], ctx[j].y * f);
        atomicAdd(&ctx_sh[h + 2], ctx[j].z * f);
        atomicAdd(&ctx_sh[h + 3], ctx[j].w * f);
    }
    __syncthreads();

    // ---- emit chunk partial ----
    const int t = threadIdx.x;
    float4* dst = (float4*)(part_ctx + ((size_t)b * NC_ + c) * H_);
    dst[t] = *(const float4*)(ctx_sh + t * 4);
    if (t == 0) {
        float ssum = 0.0f;
        #pragma unroll
        for (int i = 0; i < 8; ++i) ssum += s_sh[i];
        part_ms[((size_t)b * NC_ + c) * 2 + 0] = m_b;
        part_ms[((size_t)b * NC_ + c) * 2 + 1] = ssum;
    }
}

// ---------------------------------------------------------------------------
// Combine chunk partials -> context (into concat buffer) + per-b (m, 1/s).
// grid = B, block = 256.
// ---------------------------------------------------------------------------
__global__ __launch_bounds__(256) void k_attn_final(
    const float* __restrict__ part_ctx, const float* __restrict__ part_ms,
    float* __restrict__ concatbuf, float* __restrict__ stats)
{
    const int b = blockIdx.x, t = threadIdx.x;
    float m = -3.4e38f;
    #pragma unroll
    for (int c = 0; c < NC_; ++c)
        m = fmaxf(m, part_ms[((size_t)b * NC_ + c) * 2]);
    float s = 0.0f;
    float fc[NC_];
    #pragma unroll
    for (int c = 0; c < NC_; ++c) {
        fc[c] = __expf(part_ms[((size_t)b * NC_ + c) * 2] - m);
        s += part_ms[((size_t)b * NC_ + c) * 2 + 1] * fc[c];
    }
    const float inv = 1.0f / s;

    const int h = t * 4;
    float4 acc = make_float4(0.f, 0.f, 0.f, 0.f);
    #pragma unroll
    for (int c = 0; c < NC_; ++c) {
        const float4 v = *(const float4*)(part_ctx + ((size_t)b * NC_ + c) * H_ + h);
        acc.x += v.x * fc[c]; acc.y += v.y * fc[c];
        acc.z += v.z * fc[c]; acc.w += v.w * fc[c];
    }
    float4 r = make_float4(acc.x * inv, acc.y * inv, acc.z * inv, acc.w * inv);
    *(float4*)(concatbuf + (size_t)b * (2 * H_) + H_ + h) = r;
    if (t == 0) { stats[b * 2 + 0] = m; stats[b * 2 + 1] = inv; }
}

// ---------------------------------------------------------------------------
// Normalized attention weights from raw energies. grid (B, L/256), block 256.
// ---------------------------------------------------------------------------
__global__ __launch_bounds__(256) void k_attn_weights(
    const float* __restrict__ energies, const float* __restrict__ stats,
    float* __restrict__ attn_out)
{
    const int b = blockIdx.x;
    const int l = blockIdx.y * 256 + threadIdx.x;
    const float m = stats[b * 2 + 0], inv = stats[b * 2 + 1];
    attn_out[(size_t)b * L_ + l] =
        __expf(energies[(size_t)b * L_ + l] - m) * inv;
}

// prior -> out_in[:, H:H+P].  grid = B, block = P.
__global__ __launch_bounds__(128) void k_prior(
    const float* __restrict__ prior, float* __restrict__ out_in)
{
    const int b = blockIdx.x, p = threadIdx.x;
    out_in[(size_t)b * (H_ + P_) + H_ + p] = prior[(size_t)b * P_ + p];
}

// ---------------------------------------------------------------------------
extern "C" void kernel_launch(void* const* d_in, const int* in_sizes, int n_in,
                              void* d_out, int out_size, void* d_ws, size_t ws_size,
                              hipStream_t stream) {
    const int*   seq     = (const int*)  d_in[0];
    const float* lasth   = (const float*)d_in[1];   // [1,B,H]
    const float* enc     = (const float*)d_in[2];   // [L,B,H]
    const float* prior   = (const float*)d_in[3];   // [B,P]
    const float* emb     = (const float*)d_in[4];   // [V,H]
    const float* W_ih    = (const float*)d_in[5];   // [3H,H]
    const float* W_hh    = (const float*)d_in[6];   // [3H,H]
    const float* b_ih    = (const float*)d_in[7];
    const float* b_hh    = (const float*)d_in[8];
    const float* W_con   = (const float*)d_in[9];   // [H,2H]
    const float* b_con   = (const float*)d_in[10];
    const float* W_out   = (const float*)d_in[11];  // [V,H+P]
    const float* b_out   = (const float*)d_in[12];

    float* out = (float*)d_out;
    float* out_logits = out;                                       // [B,V]
    float* out_hidden = out + (size_t)B_ * V_;                     // [1,B,H]
    float* out_attn   = out + (size_t)B_ * V_ + (size_t)B_ * H_;   // [B,1,L]

    // Workspace layout (floats)
    float* ws       = (float*)d_ws;
    float* xbuf     = ws;                                // [B,H]
    float* gi       = xbuf     + (size_t)B_ * H_;        // [B,3H]
    float* gh       = gi       + (size_t)B_ * 3 * H_;    // [B,3H]
    float* concatb  = gh       + (size_t)B_ * 3 * H_;    // [B,2H]
    float* out_in   = concatb  + (size_t)B_ * 2 * H_;    // [B,H+P]
    float* energies = out_in   + (size_t)B_ * (H_ + P_); // [B,L]
    float* part_ctx = energies + (size_t)B_ * L_;        // [B,NC,H]
    float* part_ms  = part_ctx + (size_t)B_ * NC_ * H_;  // [B,NC,2]
    float* stats    = part_ms  + (size_t)B_ * NC_ * 2;   // [B,2]

    // 1) embedding gather
    k_embed<<<dim3(B_, H_ / 256), 256, 0, stream>>>(seq, emb, xbuf);

    // 2) GRU input/hidden GEMMs (WMMA f32): [64,1024] x [1024,3072]
    gemm_wmma_f32<<<3 * H_ / 16, 128, 0, stream>>>(xbuf, H_, W_ih, H_, b_ih,
                                                   gi, 3 * H_, H_, 0);
    gemm_wmma_f32<<<3 * H_ / 16, 128, 0, stream>>>(lasth, H_, W_hh, H_, b_hh,
                                                   gh, 3 * H_, H_, 0);

    // 3) GRU gates -> h_new (also into concat[:, :H])
    k_gru_gate<<<dim3(B_, H_ / 256), 256, 0, stream>>>(gi, gh, lasth,
                                                       out_hidden, concatb);

    // 4) fused attention: single 512 MB pass over encoder_outputs
    k_attn_fused<<<dim3(B_, NC_), 256, 0, stream>>>(out_hidden, enc,
                                                    energies, part_ctx, part_ms);

    // 5) combine partials -> context (concat[:, H:2H]) + (m, 1/s) stats
    k_attn_final<<<B_, 256, 0, stream>>>(part_ctx, part_ms, concatb, stats);

    // 6) normalized attention weights output
    k_attn_weights<<<dim3(B_, L_ / 256), 256, 0, stream>>>(energies, stats,
                                                           out_attn);

    // 7) concat projection + tanh: [64,2048] x [2048,1024] -> out_in[:, :H]
    gemm_wmma_f32<<<H_ / 16, 128, 0, stream>>>(concatb, 2 * H_, W_con, 2 * H_,
                                               b_con, out_in, H_ + P_, 2 * H_, 1);

    // 8) prior copy
    k_prior<<<B_, P_, 0, stream>>>(prior, out_in);

    // 9) output GEMM: [64,1152] x [1152,32000] (+b_out) -> logits
    gemm_wmma_f32<<<V_ / 16, 128, 0, stream>>>(out_in, H_ + P_, W_out, H_ + P_,
                                               b_out, out_logits, V_, H_ + P_, 0);
}